// A_LIF_19739669692673
// MI455X (gfx1250) — compile-verified
//
#include <hip/hip_runtime.h>
#include <stdint.h>

// LIF forward scan, [T,B,N] f32 -> spikes [T,B,N] f32.
// Memory-bound (AI ~ 0.75 FLOP/B): stream via CDNA5 async global->LDS DMA,
// 16-deep per-wave pipeline tracked with ASYNCcnt.

#define T_STEPS   100
#define NTHREADS  256
#define VEC       4
#define DEPTH     16                      // async pipeline depth (ASYNCcnt max 63)
#define STAGE_BYTES (NTHREADS * VEC * 4)  // 4096 B per stage per block

#define STR2(x) #x
#define STR(x) STR2(x)
#define DEPTH_M1 15

__global__ __launch_bounds__(NTHREADS)
void lif_async_scan(const float* __restrict__ inpt,
                    const float* __restrict__ vth_p,
                    const float* __restrict__ vdecay_p,
                    float* __restrict__ out,
                    unsigned BN)
{
    __shared__ __align__(16) unsigned char smem[DEPTH * STAGE_BYTES]; // 64 KB

    const unsigned tid   = threadIdx.x;
    const unsigned gelem = (blockIdx.x * NTHREADS + tid) * VEC; // element index in a timestep
    const unsigned gbyte = gelem * 4u;                          // byte offset in a timestep
    const unsigned step_bytes = BN * 4u;                        // 1 MiB per timestep

    const float v_th    = vth_p[0];
    const float v_decay = vdecay_p[0];
    const float inv = 1.0f / (v_th + 1e-8f);
    const float sa  = (v_th > 0.0f) ?  inv  : -inv;   // thr = fma(mem, sa, sb)
    const float sb  = (v_th > 0.0f) ? -1.0f :  1.0f;

    // Wave-relative LDS byte address of this thread's 16B slot in stage 0.
    // Low 32 bits of the shared-aperture flat address == LDS byte address.
    const unsigned lds_slot = (unsigned)(uintptr_t)(&smem[0]) + tid * (VEC * 4);
    const uint64_t in_base  = (uint64_t)(uintptr_t)inpt;

    // ---- prologue: prefetch timesteps 0..DEPTH-1 into the 16 LDS stages ----
#pragma unroll
    for (unsigned s = 0; s < DEPTH; ++s) {
        unsigned voff  = s * step_bytes + gbyte;
        unsigned laddr = lds_slot + s * STAGE_BYTES;
        asm volatile("global_load_async_to_lds_b128 %0, %1, %2"
                     :: "v"(laddr), "v"(voff), "s"(in_base) : "memory");
    }

    float m[VEC]  = {0.f, 0.f, 0.f, 0.f};  // membrane carry
    float sp[VEC] = {0.f, 0.f, 0.f, 0.f};  // spike carry

    const float4* lds_rd = (const float4*)(smem + tid * (VEC * 4)); // stage stride = NTHREADS float4s

    // ---- main loop: consume stage t, refill with t+DEPTH ----
    for (int t = 0; t < T_STEPS - DEPTH; ++t) {
        // loads complete in order: <=15 outstanding => stage t is resident
        asm volatile("s_wait_asynccnt " STR(DEPTH_M1) ::: "memory");

        float4 x = lds_rd[(unsigned)(t & (DEPTH - 1)) * NTHREADS];

        // data now in VGPRs -> safe to overwrite this stage; issue refill ASAP
        asm volatile("s_wait_dscnt 0" ::: "memory");
        {
            unsigned tp    = (unsigned)t + DEPTH;
            unsigned voff  = tp * step_bytes + gbyte;
            unsigned laddr = lds_slot + ((unsigned)t & (DEPTH - 1)) * STAGE_BYTES;
            asm volatile("global_load_async_to_lds_b128 %0, %1, %2"
                         :: "v"(laddr), "v"(voff), "s"(in_base) : "memory");
        }

        float xv[VEC] = {x.x, x.y, x.z, x.w};
#pragma unroll
        for (int j = 0; j < VEC; ++j) {
            float k   = v_decay * (1.0f - sp[j]);
            m[j]      = fmaf(k, m[j], xv[j]);
            float thr = fmaf(m[j], sa, sb);
            sp[j]     = (thr > 0.0f) ? 1.0f : 0.0f;
        }
        *(float4*)(out + (size_t)t * BN + gelem) =
            make_float4(sp[0], sp[1], sp[2], sp[3]);
    }

    // ---- tail: all remaining DEPTH stages are in flight; drain them ----
    asm volatile("s_wait_asynccnt 0" ::: "memory");
    for (int t = T_STEPS - DEPTH; t < T_STEPS; ++t) {
        float4 x = lds_rd[(unsigned)(t & (DEPTH - 1)) * NTHREADS];
        float xv[VEC] = {x.x, x.y, x.z, x.w};
#pragma unroll
        for (int j = 0; j < VEC; ++j) {
            float k   = v_decay * (1.0f - sp[j]);
            m[j]      = fmaf(k, m[j], xv[j]);
            float thr = fmaf(m[j], sa, sb);
            sp[j]     = (thr > 0.0f) ? 1.0f : 0.0f;
        }
        *(float4*)(out + (size_t)t * BN + gelem) =
            make_float4(sp[0], sp[1], sp[2], sp[3]);
    }
}

extern "C" void kernel_launch(void* const* d_in, const int* in_sizes, int n_in,
                              void* d_out, int out_size, void* d_ws, size_t ws_size,
                              hipStream_t stream) {
    (void)n_in; (void)out_size; (void)d_ws; (void)ws_size;
    const float* inpt   = (const float*)d_in[0];
    const float* vth    = (const float*)d_in[1];
    const float* vdecay = (const float*)d_in[2];
    float* out = (float*)d_out;

    const int total = in_sizes[0];            // T*B*N
    const unsigned BN = (unsigned)(total / T_STEPS); // 262144
    const int nthreads = (int)(BN / VEC);     // 65536
    const int blocks   = nthreads / NTHREADS; // 256

    lif_async_scan<<<blocks, NTHREADS, 0, stream>>>(inpt, vth, vdecay, out, BN);
}